// SceneDepthEstimatorADMM_1468878815476
// MI455X (gfx1250) — compile-verified
//
#include <hip/hip_runtime.h>
#include <math.h>

// ---------------------------------------------------------------------------
// SceneDepthEstimatorADMM for MI455X (gfx1250, wave32)
//
// Pipeline: separable 15x15 max filter -> minmax norm -> F_Nhat = FFT2(N_hat)
// -> 30 ADMM iters { shrink; FFT2(Xx-bx), FFT2(Xy-by); freq-solve; IFFT2;
//                    Bregman update; L1 convergence + device-side early stop }
// -> clip + minmax norm.
//
// 1024-pt FFT = radix-16 (WMMA f32 16x16x4) -> twiddle -> radix-16 (WMMA)
//               -> twiddle -> radix-4 (VALU).  One 4-wave workgroup per line,
//               line lives in 16KB LDS.  Global<->LDS staging uses the CDNA5
//               async-LDS path (GLOBAL_LOAD_ASYNC_TO_LDS_B32 /
//               GLOBAL_STORE_ASYNC_FROM_LDS_B32, ASYNCcnt).
//
// Workspace: 12 planes of 8*1024*1024 f32 + small reduction region
//            (~403 MB required).
// ---------------------------------------------------------------------------

typedef __attribute__((ext_vector_type(2))) float v2f;
typedef __attribute__((ext_vector_type(8))) float v8f;

#define PI2F 6.283185307179586f
#define AGc  0.3f          /* ALPHA*GAMMA */
#define SSc  8388608       /* 8*1024*1024 */
#define HWc  1048576       /* 1024*1024   */

static __device__ __forceinline__ v8f wmma4(v2f a, v2f b, v8f c) {
  // D = A(16x4) * B(4x16) + C(16x16), full fp32
  return __builtin_amdgcn_wmma_f32_16x16x4_f32(false, a, false, b, (short)0, c,
                                               false, false);
}

static __device__ __forceinline__ unsigned lds_off(const void* p) {
  // generic pointer to LDS: low 32 bits are the LDS byte offset
  return (unsigned)(uintptr_t)p;
}

static __device__ __forceinline__ void async_load_b32(unsigned lds_addr,
                                                      unsigned voff,
                                                      const float* sbase) {
  asm volatile("global_load_async_to_lds_b32 %0, %1, %2"
               :: "v"(lds_addr), "v"(voff), "s"(sbase) : "memory");
}

static __device__ __forceinline__ void async_store_b32(unsigned voff,
                                                       unsigned lds_addr,
                                                       const float* sbase) {
  asm volatile("global_store_async_from_lds_b32 %0, %1, %2"
               :: "v"(voff), "v"(lds_addr), "s"(sbase) : "memory");
}

static __device__ __forceinline__ void wait_async0() {
#if __has_builtin(__builtin_amdgcn_s_wait_asynccnt)
  __builtin_amdgcn_s_wait_asynccnt(0);
#else
  asm volatile("s_wait_asynccnt 0" ::: "memory");
#endif
}

// ------------------------------ init ---------------------------------------
__global__ void k_init(float* bx, float* by, int* done) {
  int i = blockIdx.x * blockDim.x + threadIdx.x;
  if (i == 0) *done = 0;
  if (i < SSc) { bx[i] = 0.0f; by[i] = 0.0f; }
}

// --------------------- separable max filter (k=15) --------------------------
__global__ void k_hmax(const float* __restrict__ img, float* __restrict__ tmp3) {
  int i = blockIdx.x * blockDim.x + threadIdx.x;
  if (i >= 3 * SSc) return;
  int x  = i & 1023;
  int x0 = x - 7; if (x0 < 0) x0 = 0;
  int x1 = x + 7; if (x1 > 1023) x1 = 1023;
  int rowb = i - x;
  float m = -1e30f;
  for (int xx = x0; xx <= x1; ++xx) m = fmaxf(m, img[rowb + xx]);
  tmp3[i] = m;
}

__global__ __launch_bounds__(256)
void k_vmax_pre(const float* __restrict__ tmp3, const float* __restrict__ img,
                float* __restrict__ pre, float* __restrict__ partA) {
  int y = blockIdx.x, b = blockIdx.y, t = threadIdx.x;
  int y0 = y - 7; if (y0 < 0) y0 = 0;
  int y1 = y + 7; if (y1 > 1023) y1 = 1023;
  __shared__ float smn[256], smx[256];
  float mnv = 1e30f, mxv = -1e30f;
  for (int k = 0; k < 4; ++k) {
    int x = t + 256 * k;
    float r = -1e30f, g = -1e30f, bl = -1e30f;
    for (int yy = y0; yy <= y1; ++yy) {
      size_t o = (size_t)b * 3 * HWc + (size_t)yy * 1024 + x;
      r  = fmaxf(r,  tmp3[o]);
      g  = fmaxf(g,  tmp3[o + HWc]);
      bl = fmaxf(bl, tmp3[o + 2 * HWc]);
    }
    float v = r - fmaxf(g, bl) + img[(size_t)b * 3 * HWc + (size_t)y * 1024 + x];
    pre[(size_t)b * HWc + (size_t)y * 1024 + x] = v;
    mnv = fminf(mnv, v); mxv = fmaxf(mxv, v);
  }
  smn[t] = mnv; smx[t] = mxv; __syncthreads();
  for (int off = 128; off > 0; off >>= 1) {
    if (t < off) { smn[t] = fminf(smn[t], smn[t + off]);
                   smx[t] = fmaxf(smx[t], smx[t + off]); }
    __syncthreads();
  }
  if (t == 0) { partA[(b * 1024 + y) * 2] = smn[0];
                partA[(b * 1024 + y) * 2 + 1] = smx[0]; }
}

__global__ __launch_bounds__(256)
void k_minmax_reduce(const float* __restrict__ partA, float* mn, float* mx) {
  __shared__ float smn[256], smx[256];
  int t = threadIdx.x;
  for (int b = 0; b < 8; ++b) {
    float a = 1e30f, z = -1e30f;
    for (int j = t; j < 1024; j += 256) {
      a = fminf(a, partA[(b * 1024 + j) * 2]);
      z = fmaxf(z, partA[(b * 1024 + j) * 2 + 1]);
    }
    smn[t] = a; smx[t] = z; __syncthreads();
    for (int off = 128; off > 0; off >>= 1) {
      if (t < off) { smn[t] = fminf(smn[t], smn[t + off]);
                     smx[t] = fmaxf(smx[t], smx[t + off]); }
      __syncthreads();
    }
    if (t == 0) { mn[b] = smn[0]; mx[b] = smx[0]; }
    __syncthreads();
  }
}

__global__ void k_normalize(const float* __restrict__ in, float* __restrict__ out,
                            const float* __restrict__ mn, const float* __restrict__ mx) {
  int i = blockIdx.x * blockDim.x + threadIdx.x;
  if (i >= SSc) return;
  int b = i >> 20;
  out[i] = (in[i] - mn[b]) / (mx[b] - mn[b] + 1e-6f);
}

// ------------------------- 1024-pt WMMA FFT ---------------------------------
// N = 16*16*4,  n = n1*64 + n2*4 + n3,  k = k1 + 16*k2 + 256*k3.
// Stage1: DFT16 over n1 (WMMA).  Tw1: W256^(n2*k1).
// Stage2: DFT16 over n2 (WMMA).  Tw2: W1024^(n3*(k1+16*k2)).
// Stage3: DFT4 over n3 (VALU), staged to LDS in natural order, then a
//         linear coalesced async store.   dir = -1 fwd, +1 inv.
__global__ __launch_bounds__(128)
void k_fft1024(const float* __restrict__ in_re, const float* __restrict__ in_im,
               const float* __restrict__ sub_re,
               float* __restrict__ out_re, float* __restrict__ out_im,
               int col_mode, float dir, float scale,
               const int* __restrict__ done_flag) {
  if (*done_flag) return;
  __shared__ float Are[1024], Aim[1024], Bre[1024], Bim[1024];
  const int tid  = threadIdx.x;
  const int line = blockIdx.x;

  size_t base; int estride;
  if (col_mode) { base = (size_t)(line >> 10) * HWc + (size_t)(line & 1023); estride = 1024; }
  else          { base = (size_t)line * 1024; estride = 1; }

  // ---- async global->LDS line load (ASYNCcnt path, no VGPR staging) ----
  {
    const float* srcR = in_re + base;
#pragma unroll
    for (int k = 0; k < 8; ++k) {
      int i = tid + 128 * k;
      async_load_b32(lds_off(&Are[i]), (unsigned)(i * estride) * 4u, srcR);
    }
    if (in_im) {
      const float* srcI = in_im + base;
#pragma unroll
      for (int k = 0; k < 8; ++k) {
        int i = tid + 128 * k;
        async_load_b32(lds_off(&Aim[i]), (unsigned)(i * estride) * 4u, srcI);
      }
    } else {
      for (int i = tid; i < 1024; i += 128) Aim[i] = 0.0f;
    }
    if (sub_re) {   // stage subtrahend into Bre, subtract LDS-side below
      const float* srcS = sub_re + base;
#pragma unroll
      for (int k = 0; k < 8; ++k) {
        int i = tid + 128 * k;
        async_load_b32(lds_off(&Bre[i]), (unsigned)(i * estride) * 4u, srcS);
      }
    }
    wait_async0();
  }
  __syncthreads();
  if (sub_re) {
    for (int i = tid; i < 1024; i += 128) Are[i] -= Bre[i];
    __syncthreads();
  }

  const int lane = tid & 31;
  const int wave = tid >> 5;
  const int hi   = lane >> 4;      // lane half
  const int mrow = lane & 15;

  // DFT-16 matrix operand, A-layout chunks (shared by both radix-16 stages)
  v2f awr[4], awi[4];
#pragma unroll
  for (int kc = 0; kc < 4; ++kc) {
    int c0 = kc * 4 + hi * 2;
    float s0, c0f, s1, c1f;
    __sincosf(dir * (PI2F / 16.0f) * (float)(mrow * c0),       &s0, &c0f);
    __sincosf(dir * (PI2F / 16.0f) * (float)(mrow * (c0 + 1)), &s1, &c1f);
    v2f wr; wr[0] = c0f; wr[1] = c1f;
    v2f wi; wi[0] = s0;  wi[1] = s1;
    awr[kc] = wr; awi[kc] = wi;
  }

  // -------- Stage 1: wave handles n3 = wave; Y[k1][n2] = sum_n1 W16 * x
  {
    const int n2 = mrow;
    v2f xr[4], xi[4];
#pragma unroll
    for (int kc = 0; kc < 4; ++kc) {
      int r0 = kc * 4 + hi * 2;
      int a0 = r0 * 64 + n2 * 4 + wave;
      v2f tr; tr[0] = Are[a0]; tr[1] = Are[a0 + 64];
      xr[kc] = tr;
      if (in_im) { v2f ti; ti[0] = Aim[a0]; ti[1] = Aim[a0 + 64]; xi[kc] = ti; }
    }
    v8f yr = {0.f,0.f,0.f,0.f,0.f,0.f,0.f,0.f};
    v8f yi = {0.f,0.f,0.f,0.f,0.f,0.f,0.f,0.f};
#pragma unroll
    for (int kc = 0; kc < 4; ++kc) yr = wmma4(awr[kc], xr[kc], yr);
#pragma unroll
    for (int kc = 0; kc < 4; ++kc) yi = wmma4(awi[kc], xr[kc], yi);
    if (in_im) {   // uniform branch: EXEC stays all-ones for WMMA
#pragma unroll
      for (int kc = 0; kc < 4; ++kc) { v2f nw = -awi[kc]; yr = wmma4(nw, xi[kc], yr); }
#pragma unroll
      for (int kc = 0; kc < 4; ++kc) yi = wmma4(awr[kc], xi[kc], yi);
    }
    // Twiddle W256^(n2*k1), store to B at [k1][n2][n3]
#pragma unroll
    for (int r = 0; r < 8; ++r) {
      int k1 = r + 8 * hi;
      float s, c;
      __sincosf(dir * (PI2F / 256.0f) * (float)(k1 * n2), &s, &c);
      float zr = yr[r] * c - yi[r] * s;
      float zi = yr[r] * s + yi[r] * c;
      int ad = k1 * 64 + n2 * 4 + wave;
      Bre[ad] = zr; Bim[ad] = zi;
    }
  }
  __syncthreads();

  // -------- Stage 2: cols c=(k1,n3), wave tile = [16w,16w+16)
  {
    const int cg = 16 * wave + mrow;
    const int k1 = cg >> 2;
    const int n3 = cg & 3;
    v2f br2[4], bi2[4];
#pragma unroll
    for (int kc = 0; kc < 4; ++kc) {
      int r0 = kc * 4 + hi * 2;               // n2 rows
      int a0 = k1 * 64 + r0 * 4 + n3;
      v2f tr; tr[0] = Bre[a0]; tr[1] = Bre[a0 + 4];
      v2f ti; ti[0] = Bim[a0]; ti[1] = Bim[a0 + 4];
      br2[kc] = tr; bi2[kc] = ti;
    }
    v8f zr = {0.f,0.f,0.f,0.f,0.f,0.f,0.f,0.f};
    v8f zi = {0.f,0.f,0.f,0.f,0.f,0.f,0.f,0.f};
#pragma unroll
    for (int kc = 0; kc < 4; ++kc) zr = wmma4(awr[kc], br2[kc], zr);
#pragma unroll
    for (int kc = 0; kc < 4; ++kc) { v2f nw = -awi[kc]; zr = wmma4(nw, bi2[kc], zr); }
#pragma unroll
    for (int kc = 0; kc < 4; ++kc) zi = wmma4(awi[kc], br2[kc], zi);
#pragma unroll
    for (int kc = 0; kc < 4; ++kc) zi = wmma4(awr[kc], bi2[kc], zi);
    // Twiddle W1024^(n3*(k1+16*k2)), store to A at [(k1*16+k2)*4 + n3]
#pragma unroll
    for (int r = 0; r < 8; ++r) {
      int k2 = r + 8 * hi;
      float s, c;
      __sincosf(dir * (PI2F / 1024.0f) * (float)(n3 * (k1 + 16 * k2)), &s, &c);
      float gr = zr[r] * c - zi[r] * s;
      float gi = zr[r] * s + zi[r] * c;
      int ad = ((k1 << 4) + k2) * 4 + n3;
      Are[ad] = gr; Aim[ad] = gi;
    }
  }
  __syncthreads();

  // -------- Stage 3: radix-4 over n3; stage to LDS in natural k order
  const float sg = dir;
  for (int g = tid; g < 256; g += 128) {
    int b4 = g * 4;
    float v0r = Are[b4],     v0i = Aim[b4];
    float v1r = Are[b4 + 1], v1i = Aim[b4 + 1];
    float v2r = Are[b4 + 2], v2i = Aim[b4 + 2];
    float v3r = Are[b4 + 3], v3i = Aim[b4 + 3];
    float t0r = v0r + v2r, t0i = v0i + v2i;
    float t1r = v0r - v2r, t1i = v0i - v2i;
    float t2r = v1r + v3r, t2i = v1i + v3i;
    float t3r = v1r - v3r, t3i = v1i - v3i;
    int k1 = g >> 4, k2 = g & 15;
    int kb = k1 + 16 * k2;            // digit-swap: bijection over [0,256)
    Bre[kb]       = scale * (t0r + t2r);
    Bim[kb]       = scale * (t0i + t2i);
    Bre[kb + 256] = scale * (t1r - sg * t3i);
    Bim[kb + 256] = scale * (t1i + sg * t3r);
    Bre[kb + 512] = scale * (t0r - t2r);
    Bim[kb + 512] = scale * (t0i - t2i);
    Bre[kb + 768] = scale * (t1r + sg * t3i);
    Bim[kb + 768] = scale * (t1i - sg * t3r);
  }
  __syncthreads();

  // -------- linear coalesced async LDS->global store (drained by s_wait_asynccnt 0)
  {
    const float* dstR = out_re + base;
#pragma unroll
    for (int k = 0; k < 8; ++k) {
      int i = tid + 128 * k;
      async_store_b32((unsigned)(i * estride) * 4u, lds_off(&Bre[i]), dstR);
    }
    if (out_im) {
      const float* dstI = out_im + base;
#pragma unroll
      for (int k = 0; k < 8; ++k) {
        int i = tid + 128 * k;
        async_store_b32((unsigned)(i * estride) * 4u, lds_off(&Bim[i]), dstI);
      }
    }
    wait_async0();
  }
}

// ------------------------- per-iteration pointwise ---------------------------
__global__ void k_pointwise_X(const float* __restrict__ N,
                              const float* __restrict__ bx, const float* __restrict__ by,
                              float* __restrict__ Xx, float* __restrict__ Xy,
                              const int* __restrict__ done) {
  if (*done) return;
  int i = blockIdx.x * blockDim.x + threadIdx.x;
  if (i >= SSc) return;
  int x = i & 1023, y = (i >> 10) & 1023;
  int ix = (x == 1023) ? i - 1023 : i + 1;
  int iy = (y == 1023) ? i - 1023 * 1024 : i + 1024;
  float n = N[i];
  float dnx = N[ix] - n, dny = N[iy] - n;
  const float inv_sig = 1.0f / (0.05f + 1e-12f);
  float wx = 1.0f + 0.5f * __expf(-fabsf(dnx) * inv_sig);
  float wy = 1.0f + 0.5f * __expf(-fabsf(dny) * inv_sig);
  float vx = dnx + bx[i], vy = dny + by[i];
  Xx[i] = copysignf(fmaxf(fabsf(vx) - wx * 0.25f, 0.0f), vx);   // Wx/(2*GAMMA)
  Xy[i] = copysignf(fmaxf(fabsf(vy) - wy * 0.25f, 0.0f), vy);
}

__global__ void k_combine(float* __restrict__ AR, float* __restrict__ AI,
                          const float* __restrict__ BR, const float* __restrict__ BI,
                          const float* __restrict__ FNR, const float* __restrict__ FNI,
                          const int* __restrict__ done) {
  if (*done) return;
  int i = blockIdx.x * blockDim.x + threadIdx.x;
  if (i >= SSc) return;
  int kx = i & 1023, ky = (i >> 10) & 1023;
  float sx, cx, sy, cy;
  __sincosf(PI2F * (float)kx * (1.0f / 1024.0f), &sx, &cx);
  __sincosf(PI2F * (float)ky * (1.0f / 1024.0f), &sy, &cy);
  float cdxr = 1.0f - cx, cdxi = sx;    // conj(otf_dx)
  float cdyr = 1.0f - cy, cdyi = sy;    // conj(otf_dy)
  float den = 1.0f + AGc * ((2.0f - 2.0f * cx) + (2.0f - 2.0f * cy));
  float ar = AR[i], ai = AI[i], br = BR[i], bi = BI[i];
  float nr = FNR[i] + AGc * (cdxr * ar - cdxi * ai + cdyr * br - cdyi * bi);
  float ni = FNI[i] + AGc * (cdxr * ai + cdxi * ar + cdyr * bi + cdyi * br);
  float id = 1.0f / den;
  AR[i] = nr * id; AI[i] = ni * id;
}

__global__ __launch_bounds__(256)
void k_update(const float* __restrict__ N, float* __restrict__ N2,
              float* __restrict__ bx, float* __restrict__ by,
              const float* __restrict__ Xx, const float* __restrict__ Xy,
              float* __restrict__ partB, const int* __restrict__ done) {
  int y = blockIdx.x, b = blockIdx.y, t = threadIdx.x;
  int rowbase = (b * 1024 + y) * 1024;
  if (*done) {   // keep next-N == N so host-side buffer swap stays consistent
    for (int k = 0; k < 4; ++k) { int x = t + 256 * k; N2[rowbase + x] = N[rowbase + x]; }
    return;
  }
  __shared__ float sn[256], sd[256];
  float an = 0.0f, ad = 0.0f;
  for (int k = 0; k < 4; ++k) {
    int x = t + 256 * k;
    int i = rowbase + x;
    int ix = (x == 1023) ? i - 1023 : i + 1;
    int iy = (y == 1023) ? i - 1023 * 1024 : i + 1024;
    float n2 = N2[i];
    float dnx2 = N2[ix] - n2, dny2 = N2[iy] - n2;
    bx[i] += dnx2 - Xx[i];
    by[i] += dny2 - Xy[i];
    float n = N[i];
    an += fabsf(n2 - n); ad += fabsf(n);
  }
  sn[t] = an; sd[t] = ad; __syncthreads();
  for (int off = 128; off > 0; off >>= 1) {
    if (t < off) { sn[t] += sn[t + off]; sd[t] += sd[t + off]; }
    __syncthreads();
  }
  if (t == 0) { partB[(b * 1024 + y) * 2] = sn[0];
                partB[(b * 1024 + y) * 2 + 1] = sd[0]; }
}

__global__ __launch_bounds__(256)
void k_eps(const float* __restrict__ partB, int* done) {
  if (*done) return;
  __shared__ float sn[256], sd[256];
  int t = threadIdx.x;
  float maxeps = 0.0f;
  for (int b = 0; b < 8; ++b) {
    float n = 0.0f, d = 0.0f;
    for (int j = t; j < 1024; j += 256) {
      n += partB[(b * 1024 + j) * 2];
      d += partB[(b * 1024 + j) * 2 + 1];
    }
    sn[t] = n; sd[t] = d; __syncthreads();
    for (int off = 128; off > 0; off >>= 1) {
      if (t < off) { sn[t] += sn[t + off]; sd[t] += sd[t + off]; }
      __syncthreads();
    }
    if (t == 0) { float e = sn[0] / (sd[0] + 1e-8f); if (e > maxeps) maxeps = e; }
    __syncthreads();
  }
  if (t == 0 && maxeps < 1e-4f) *done = 1;
}

__global__ __launch_bounds__(256)
void k_clip_minmax(const float* __restrict__ N, float* __restrict__ C,
                   float* __restrict__ partA) {
  int y = blockIdx.x, b = blockIdx.y, t = threadIdx.x;
  __shared__ float smn[256], smx[256];
  float mnv = 1e30f, mxv = -1e30f;
  int rowbase = (b * 1024 + y) * 1024;
  for (int k = 0; k < 4; ++k) {
    int i = rowbase + t + 256 * k;
    float v = fminf(fmaxf(N[i], 0.0f), 1.0f);
    C[i] = v;
    mnv = fminf(mnv, v); mxv = fmaxf(mxv, v);
  }
  smn[t] = mnv; smx[t] = mxv; __syncthreads();
  for (int off = 128; off > 0; off >>= 1) {
    if (t < off) { smn[t] = fminf(smn[t], smn[t + off]);
                   smx[t] = fmaxf(smx[t], smx[t + off]); }
    __syncthreads();
  }
  if (t == 0) { partA[(b * 1024 + y) * 2] = smn[0];
                partA[(b * 1024 + y) * 2 + 1] = smx[0]; }
}

// ------------------------------ host orchestration ---------------------------
extern "C" void kernel_launch(void* const* d_in, const int* in_sizes, int n_in,
                              void* d_out, int out_size, void* d_ws, size_t ws_size,
                              hipStream_t stream) {
  (void)in_sizes; (void)n_in; (void)out_size; (void)ws_size;
  const float* img = (const float*)d_in[0];
  float* f = (float*)d_ws;
  const size_t S = (size_t)SSc;
  float *N   = f,          *N2  = f + S,      *bx  = f + 2 * S, *by  = f + 3 * S;
  float *Xx  = f + 4 * S,  *Xy  = f + 5 * S;
  float *AR  = f + 6 * S,  *AI  = f + 7 * S,  *BR  = f + 8 * S, *BI  = f + 9 * S;
  float *FNR = f + 10 * S, *FNI = f + 11 * S;
  float *red = f + 12 * S;                 // needs ~403 MB of workspace total
  int*   done  = (int*)red;
  float* mn    = red + 8;
  float* mx    = red + 16;
  float* partA = red + 32;                 // 8*1024*2
  float* partB = partA + 8 * 1024 * 2;     // 8*1024*2
  float* tmp3  = Xx;                       // setup-time alias over Xx,Xy,AR
  float* Cbuf  = Xx;                       // final clip scratch

  dim3 b256(256);
  dim3 gElem((SSc + 255) / 256);
  dim3 gRow(1024, 8);
  dim3 gFFT(8192);     // 8 batches * 1024 lines
  dim3 bFFT(128);      // 4 wave32 -> one 1024-pt line

  k_init<<<gElem, b256, 0, stream>>>(bx, by, done);

  // ---- prior: separable max filter + D_mip + Ir, minmax-normalize to N_hat
  k_hmax<<<dim3((3 * SSc + 255) / 256), b256, 0, stream>>>(img, tmp3);
  k_vmax_pre<<<gRow, b256, 0, stream>>>(tmp3, img, N, partA);
  k_minmax_reduce<<<1, 256, 0, stream>>>(partA, mn, mx);
  k_normalize<<<gElem, b256, 0, stream>>>(N, N, mn, mx);

  // ---- F_Nhat = FFT2(N_hat)
  k_fft1024<<<gFFT, bFFT, 0, stream>>>(N, nullptr, nullptr, FNR, FNI, 0, -1.f, 1.f, done);
  k_fft1024<<<gFFT, bFFT, 0, stream>>>(FNR, FNI, nullptr, FNR, FNI, 1, -1.f, 1.f, done);

  float* curN = N;
  float* nxtN = N2;
  for (int it = 0; it < 30; ++it) {
    k_pointwise_X<<<gElem, b256, 0, stream>>>(curN, bx, by, Xx, Xy, done);
    // Fx = FFT2(Xx - bx)   (subtraction fused into the row-pass LDS load)
    k_fft1024<<<gFFT, bFFT, 0, stream>>>(Xx, nullptr, bx, AR, AI, 0, -1.f, 1.f, done);
    k_fft1024<<<gFFT, bFFT, 0, stream>>>(AR, AI, nullptr, AR, AI, 1, -1.f, 1.f, done);
    // Fy = FFT2(Xy - by)
    k_fft1024<<<gFFT, bFFT, 0, stream>>>(Xy, nullptr, by, BR, BI, 0, -1.f, 1.f, done);
    k_fft1024<<<gFFT, bFFT, 0, stream>>>(BR, BI, nullptr, BR, BI, 1, -1.f, 1.f, done);
    // G = (F_Nhat + a*g*(conj(dx)Fx + conj(dy)Fy)) / denom  -> AR/AI
    k_combine<<<gElem, b256, 0, stream>>>(AR, AI, BR, BI, FNR, FNI, done);
    // N_new = real(IFFT2(G))   (1/1024 per inverse pass -> 1/(H*W) total)
    k_fft1024<<<gFFT, bFFT, 0, stream>>>(AR, AI, nullptr, AR, AI, 1, +1.f,
                                         1.0f / 1024.0f, done);
    k_fft1024<<<gFFT, bFFT, 0, stream>>>(AR, AI, nullptr, nxtN, nullptr, 0, +1.f,
                                         1.0f / 1024.0f, done);
    // Bregman update + convergence partials; early-stop flag
    k_update<<<gRow, b256, 0, stream>>>(curN, nxtN, bx, by, Xx, Xy, partB, done);
    k_eps<<<1, 256, 0, stream>>>(partB, done);
    float* tsw = curN; curN = nxtN; nxtN = tsw;
  }

  // ---- clip + final minmax normalization -> d_out
  k_clip_minmax<<<gRow, b256, 0, stream>>>(curN, Cbuf, partA);
  k_minmax_reduce<<<1, 256, 0, stream>>>(partA, mn, mx);
  k_normalize<<<gElem, b256, 0, stream>>>(Cbuf, (float*)d_out, mn, mx);
}